// Model_70746701300307
// MI455X (gfx1250) — compile-verified
//
#include <hip/hip_runtime.h>
#include <hip/hip_bf16.h>

typedef __attribute__((ext_vector_type(2))) float v2f;
typedef __attribute__((ext_vector_type(8))) float v8f;

#define NNODES 100000
#define NEDGES 1600000
#define F_IN   128
#define HID    64

// ---------------------------------------------------------------------------
// GEMM: out[N,64] = X[N,128] @ W[64,128]^T + bias, via V_WMMA_F32_16X16X4_F32.
// Block = 256 threads = 8 waves. Wave w: row tile = (blockIdx.x*2 + (w>>2))*16,
// h tile = (w&3)*16. Each wave produces one 16x16 output tile, accumulating
// K=128 in 32 chained WMMA steps.
//
// A (16x4 f32) lane layout: lane L holds A[M = L&15, K = 2*(L>>4) + vgpr].
// B (4x16 f32) lane layout (mirror): lane L holds B[K = 2*(L>>4) + vgpr, N = L&15]
//   and B[k,n] = W[hBase+n, k], so A and B fragment loads share index math.
// C/D (16x16 f32): vgpr v, lane L -> D[M = v + 8*(L>>4), N = L&15].
// ---------------------------------------------------------------------------
__global__ void k_gemm_wmma(const float* __restrict__ X,
                            const float* __restrict__ W,
                            const float* __restrict__ bias,
                            float* __restrict__ out) {
    const int lane = threadIdx.x & 31;
    const int wv   = threadIdx.x >> 5;
    const int rowBase = (blockIdx.x * 2 + (wv >> 2)) * 16;
    const int hBase   = (wv & 3) * 16;
    const int m    = lane & 15;     // row within A tile / row within W tile / N col
    const int half = lane >> 4;     // K half selector

    const float* __restrict__ xr = X + (size_t)(rowBase + m) * F_IN;
    const float* __restrict__ wr = W + (size_t)(hBase  + m) * F_IN;

    v8f acc = {};
#pragma unroll 8
    for (int k = 0; k < F_IN; k += 4) {
        v2f a, b;
        a.x = xr[k + 2 * half];
        a.y = xr[k + 2 * half + 1];
        b.x = wr[k + 2 * half];
        b.y = wr[k + 2 * half + 1];
        acc = __builtin_amdgcn_wmma_f32_16x16x4_f32(
            /*neg_a=*/false, a, /*neg_b=*/false, b,
            /*c_mod=*/(short)0, acc, /*reuse_a=*/false, /*reuse_b=*/false);
    }

    const float bcol = bias[hBase + m];   // m == N column index for this lane
#pragma unroll
    for (int v = 0; v < 8; ++v) {
        const int r = rowBase + v + 8 * half;
        out[(size_t)r * HID + hBase + m] = acc[v] + bcol;
    }
}

// ---------------------------------------------------------------------------
__global__ void k_zero(float* __restrict__ p, int n) {
    int i = blockIdx.x * 256 + threadIdx.x;
    if (i < n) p[i] = 0.0f;
}

__global__ void k_hist(const int* __restrict__ col, float* __restrict__ cnt, int e) {
    int i = blockIdx.x * 256 + threadIdx.x;
    if (i < e) atomicAdd(&cnt[col[i]], 1.0f);
}

__global__ void k_dinv(const float* __restrict__ cntPri,
                       const float* __restrict__ cntSup,
                       float* __restrict__ dPri03,
                       float* __restrict__ dPri05,
                       float* __restrict__ dSup05, int n) {
    int i = blockIdx.x * 256 + threadIdx.x;
    if (i < n) {
        float a = cntPri[i];
        float b = cntSup[i];
        dPri03[i] = rsqrtf(a + 0.3f);
        dPri05[i] = rsqrtf(a + 0.5f);
        dSup05[i] = rsqrtf(b + 0.5f);
    }
}

// h_out[i,:] (=| +=) ((1-eps) + eps*dinv[i]^2) * h_in[i,:]
__global__ void k_self(const float* __restrict__ hin,
                       const float* __restrict__ dinv,
                       float eps, float* __restrict__ hout,
                       int total, int accum) {
    int i = blockIdx.x * 256 + threadIdx.x;
    if (i >= total) return;
    int node = i >> 6;                    // HID == 64
    float di = dinv[node];
    float coef = (1.0f - eps) + eps * di * di;
    float v = coef * hin[i];
    if (accum) hout[i] += v;
    else       hout[i]  = v;
}

// For each edge (r -> c): hout[c,:] += dinv[r]*dinv[c] * hin[r,:]
// 16 lanes per edge, one float4 per lane (64 features).
__global__ void k_scatter(const int* __restrict__ row,
                          const int* __restrict__ col,
                          const float* __restrict__ dinv,
                          const float* __restrict__ hin,
                          float* __restrict__ hout, int e) {
    int idx = blockIdx.x * 256 + threadIdx.x;
    int ed  = idx >> 4;
    if (ed >= e) return;
    int f = (idx & 15) << 2;
    int r = row[ed];
    int c = col[ed];
    float nrm = dinv[r] * dinv[c];
    const float4* __restrict__ src =
        (const float4*)(hin + (size_t)r * HID + f);
    float4 v = *src;
    float* dst = hout + (size_t)c * HID + f;
    atomicAdd(dst + 0, nrm * v.x);
    atomicAdd(dst + 1, nrm * v.y);
    atomicAdd(dst + 2, nrm * v.z);
    atomicAdd(dst + 3, nrm * v.w);
}

// ---------------------------------------------------------------------------
extern "C" void kernel_launch(void* const* d_in, const int* in_sizes, int n_in,
                              void* d_out, int out_size, void* d_ws, size_t ws_size,
                              hipStream_t stream) {
    const float* x    = (const float*)d_in[0];
    const int*   priE = (const int*)  d_in[1];   // [2, E]: rows then cols
    const int*   supE = (const int*)  d_in[2];
    const float* w1   = (const float*)d_in[3];
    const float* b1   = (const float*)d_in[4];
    const float* w2   = (const float*)d_in[5];
    const float* b2   = (const float*)d_in[6];

    float* out = (float*)d_out;
    float* ws  = (float*)d_ws;

    const int N = NNODES, E = NEDGES;
    const float EPS1 = 0.3f, EPS2 = 0.5f;

    float* cntPri = ws;                       // N
    float* cntSup = ws + (size_t)N;           // N
    float* dPri03 = ws + (size_t)2 * N;       // N
    float* dPri05 = ws + (size_t)3 * N;       // N
    float* dSup05 = ws + (size_t)4 * N;       // N
    float* bufA   = ws + (size_t)5 * N;       // N*64
    float* bufB   = bufA + (size_t)N * HID;   // N*64

    float* z1 = out;                          // [N,64]
    float* z2 = out + (size_t)N * HID;        // [N,64]

    const int total  = N * HID;               // 6,400,000
    const int gSelf  = total / 256;           // 25000 exact
    const int gScat  = (E * 16) / 256;        // 100000 exact
    const int gHist  = E / 256;               // 6250 exact
    const int gGemm  = N / 32;                // 3125 exact (32 rows/block)
    const int gNode  = (N + 255) / 256;
    const int gZero  = (2 * N + 255) / 256;

    // --- degree counts + normalization factors ---
    k_zero<<<gZero, 256, 0, stream>>>(cntPri, 2 * N);   // cntPri+cntSup contiguous
    k_hist<<<gHist, 256, 0, stream>>>(priE + E, cntPri, E);
    k_hist<<<gHist, 256, 0, stream>>>(supE + E, cntSup, E);
    k_dinv<<<gNode, 256, 0, stream>>>(cntPri, cntSup, dPri03, dPri05, dSup05, N);

    // --- z1 = aspsgc(x, w1, b1, pri, eps=0.3, K=2) ---
    k_gemm_wmma<<<gGemm, 256, 0, stream>>>(x, w1, b1, bufA);
    k_self   <<<gSelf, 256, 0, stream>>>(bufA, dPri03, EPS1, bufB, total, 0);
    k_scatter<<<gScat, 256, 0, stream>>>(priE, priE + E, dPri03, bufA, bufB, E);
    k_self   <<<gSelf, 256, 0, stream>>>(bufB, dPri03, EPS1, z1, total, 0);
    k_scatter<<<gScat, 256, 0, stream>>>(priE, priE + E, dPri03, bufB, z1, E);

    // --- z2 = aspsgc(x, w2, b2, sup, eps=0.5, K=2) ---
    k_gemm_wmma<<<gGemm, 256, 0, stream>>>(x, w2, b2, bufA);
    k_self   <<<gSelf, 256, 0, stream>>>(bufA, dSup05, EPS2, bufB, total, 0);
    k_scatter<<<gScat, 256, 0, stream>>>(supE, supE + E, dSup05, bufA, bufB, E);
    k_self   <<<gSelf, 256, 0, stream>>>(bufB, dSup05, EPS2, z2, total, 0);
    k_scatter<<<gScat, 256, 0, stream>>>(supE, supE + E, dSup05, bufB, z2, E);

    // --- z2 += aspsgc(x, w2, b2, pri, eps=0.5, K=2) ---
    k_gemm_wmma<<<gGemm, 256, 0, stream>>>(x, w2, b2, bufA);
    k_self   <<<gSelf, 256, 0, stream>>>(bufA, dPri05, EPS2, bufB, total, 0);
    k_scatter<<<gScat, 256, 0, stream>>>(priE, priE + E, dPri05, bufA, bufB, E);
    k_self   <<<gSelf, 256, 0, stream>>>(bufB, dPri05, EPS2, z2, total, 1);  // accumulate
    k_scatter<<<gScat, 256, 0, stream>>>(priE, priE + E, dPri05, bufB, z2, E);
}